// SimpleHashEncoder1D_33603824124489
// MI455X (gfx1250) — compile-verified
//
#include <hip/hip_runtime.h>
#include <stdint.h>

// SimpleHashEncoder1D for MI455X (gfx1250).
// L=16 levels, F=2 feats, T=2^19 entries, N=2^21 points.
// b == 2 exactly => scales[l] = 16*2^l - 1, level-l indices < 16*2^l.
// Levels 0..9 touch only the first 8192 entries (64KB) -> staged in LDS.
// Staging split: first 32KB via TDM tensor_load_to_lds (TENSORcnt),
// second 32KB via per-lane global_load_async_to_lds_b128 (ASYNCcnt).
// Gathers via explicit addrspace(3)/(1) pointers -> ds_load_b64 /
// global_load_b64 (no flat merging). Output is the bandwidth limiter
// (256MB stores); (point, level-pair) decomposition gives one coalesced
// 16B global_store_b128 per lane.

#define ENC_L        16
#define ENC_T        524288      // 2^19
#define ENC_NPTS     2097152     // 2^21
#define STAGE_ENT    8192        // f2 entries staged (levels 0..9), 64KB
#define TDM_ENT      4096        // f2 entries staged via TDM (32KB)
#define BLOCK        512
#define GRID         1024

// Native vector types (no C++ class machinery -> work through any addrspace).
typedef float        nf2 __attribute__((ext_vector_type(2)));
typedef float        nf4 __attribute__((ext_vector_type(4)));
typedef int          v4i __attribute__((vector_size(16)));   // async builtin pointee
typedef unsigned int v4u_t __attribute__((ext_vector_type(4)));
typedef int          v4i_t __attribute__((ext_vector_type(4)));
typedef int          v8i_t __attribute__((ext_vector_type(8)));

// Address-space-qualified gather pointers (prevents flat-load merging).
typedef const __attribute__((address_space(3))) nf2* lds_f2p;
typedef const __attribute__((address_space(1))) nf2* glb_f2p;

__global__ __launch_bounds__(BLOCK)
void hashenc1d_kernel(const float* __restrict__ x,
                      const float* __restrict__ ht,
                      const int*   __restrict__ boundp,
                      float*       __restrict__ out)
{
    __shared__ nf2 s_tab[STAGE_ENT];   // 64 KB LDS

    const int tid = threadIdx.x;
    const nf2* __restrict__ ht2 = (const nf2*)ht;

    // ---- stage hot table region (levels 0..9) into LDS ----------------
#if __has_builtin(__builtin_amdgcn_tensor_load_to_lds) && \
    __has_builtin(__builtin_amdgcn_global_load_async_to_lds_b128)
    // (a) TDM DMA: entries [0, TDM_ENT) = 32KB, issued once by wave 0.
    //     1D tile: data_size=4B, tensor_dim0 = tile_dim0 = 8192 floats.
    if (tid == 0) {
        unsigned long long ga = (unsigned long long)(uintptr_t)ht;
        unsigned int lds_off =
            (unsigned int)(uintptr_t)(lds_f2p)&s_tab[0];   // as(3) byte offset
        // D# group0: count=1 | lds_addr | global_addr[56:0] | type=2
        v4u_t g0 = { 1u,
                     lds_off,
                     (unsigned int)ga,
                     (unsigned int)((ga >> 32) & 0x01FFFFFFu) | (2u << 30) };
        // D# group1: data_size=2(4B); tensor_dim0=8192; tensor_dim1=1;
        //            tile_dim0=8192; tile_dim1/2=0; dim0_stride=8192.
        v8i_t g1 = { (int)0x00020000,   // wg_mask=0, data_size=4B
                     (int)0x20000000,   // tensor_dim0[15:0]=8192 in bits[63:48]
                     (int)0x00010000,   // tensor_dim0 hi=0, tensor_dim1=1
                     (int)0x20000000,   // tile_dim0=8192 in bits[127:112]
                     0,                 // tile_dim1=0, tile_dim2=0 (1D tile)
                     8192,              // tensor_dim0_stride[31:0]
                     0, 0 };
        v4i_t g2 = { 0, 0, 0, 0 };
        v4i_t g3 = { 0, 0, 0, 0 };
        v8i_t g4 = { 0, 0, 0, 0, 0, 0, 0, 0 };
        __builtin_amdgcn_tensor_load_to_lds(g0, g1, g2, g3, g4, 0);
#if __has_builtin(__builtin_amdgcn_s_wait_tensorcnt)
        __builtin_amdgcn_s_wait_tensorcnt(0);
#else
        asm volatile("s_wait_tensorcnt 0x0" ::: "memory");
#endif
    }
    // (b) per-lane async copies: entries [TDM_ENT, STAGE_ENT) = 32KB.
    for (int c = TDM_ENT / 2 + tid; c < STAGE_ENT / 2; c += BLOCK) {
        __builtin_amdgcn_global_load_async_to_lds_b128(
            (__attribute__((address_space(1))) v4i*)(ht2 + 2 * c),
            (__attribute__((address_space(3))) v4i*)(&s_tab[2 * c]),
            0, 0);
    }
#if __has_builtin(__builtin_amdgcn_s_wait_asynccnt)
    __builtin_amdgcn_s_wait_asynccnt(0);
#else
    asm volatile("s_wait_asynccnt 0" ::: "memory");
#endif
#elif __has_builtin(__builtin_amdgcn_global_load_async_to_lds_b128)
    for (int c = tid; c < STAGE_ENT / 2; c += BLOCK) {
        __builtin_amdgcn_global_load_async_to_lds_b128(
            (__attribute__((address_space(1))) v4i*)(ht2 + 2 * c),
            (__attribute__((address_space(3))) v4i*)(&s_tab[2 * c]),
            0, 0);
    }
    asm volatile("s_wait_asynccnt 0" ::: "memory");
#else
    for (int c = tid; c < STAGE_ENT; c += BLOCK) {
        s_tab[c] = ht2[c];
    }
#endif
    __syncthreads();

    // Explicit address-space views for the gather phase.
    const lds_f2p sp = (lds_f2p)&s_tab[0];
    const glb_f2p gp = (glb_f2p)ht2;

    const float bf  = (float)boundp[0];
    const float inv = 0.5f / bf;             // xn = (x + b) / (2b)

    const int W      = ENC_NPTS * 8;         // 8 level-pairs per point
    const int stride = GRID * BLOCK;

    for (int w = blockIdx.x * BLOCK + tid; w < W; w += stride) {
        const int p = w >> 3;                // point id
        const int j = w & 7;                 // level-pair id: levels 2j, 2j+1

        // Prefetch next tranche of x (global_prefetch_b8), bounds-guarded.
        if (w + stride < W) {
            __builtin_prefetch(&x[(w + stride) >> 3], 0, 1);
        }

        const float xn = (x[p] + bf) * inv;

        const int l0 = 2 * j;
        const int l1 = 2 * j + 1;
        const float s0 = (float)(16 << l0) - 1.0f;   // 16*2^l - 1, exact
        const float s1 = (float)(16 << l1) - 1.0f;

        int i0 = (int)floorf(xn * s0 + 0.5f) & (ENC_T - 1);
        int i1 = (int)floorf(xn * s1 + 0.5f) & (ENC_T - 1);

        nf2 a, b;
        if (j <= 4) {            // levels <= 9: indices < 8192, LDS-resident
            a = sp[i0];          // ds_load_b64
            b = sp[i1];
        } else {                 // fine levels: L2-resident 4MB table
            a = gp[i0];          // global_load_b64
            b = gp[i1];
        }

        nf4 o;                   // one coalesced 16B store per lane
        o.x = a.x; o.y = a.y; o.z = b.x; o.w = b.y;
        ((nf4*)out)[w] = o;
    }
}

extern "C" void kernel_launch(void* const* d_in, const int* in_sizes, int n_in,
                              void* d_out, int out_size, void* d_ws, size_t ws_size,
                              hipStream_t stream) {
    (void)in_sizes; (void)n_in; (void)out_size; (void)d_ws; (void)ws_size;
    const float* x      = (const float*)d_in[0];
    const float* ht     = (const float*)d_in[1];
    const int*   boundp = (const int*)d_in[2];
    float*       out    = (float*)d_out;

    hipLaunchKernelGGL(hashenc1d_kernel, dim3(GRID), dim3(BLOCK), 0, stream,
                       x, ht, boundp, out);
}